// soliSpdNet_42365557408044
// MI455X (gfx1250) — compile-verified
//
#include <hip/hip_runtime.h>
#include <hip/hip_bf16.h>

typedef float v2f __attribute__((ext_vector_type(2)));
typedef float v8f __attribute__((ext_vector_type(8)));
typedef unsigned int u32x4 __attribute__((ext_vector_type(4)));
typedef int i32x4 __attribute__((ext_vector_type(4)));
typedef int i32x8 __attribute__((ext_vector_type(8)));

#define REEIG_EPS 1e-4f
#define NSWEEP 8

// ---- CDNA5 TDM availability -------------------------------------------------
#if defined(__has_builtin)
#  if __has_builtin(__builtin_amdgcn_tensor_load_to_lds) && \
      __has_builtin(__builtin_amdgcn_s_wait_tensorcnt)
#    define HAVE_TDM 1
#  else
#    define HAVE_TDM 0
#  endif
#else
#  define HAVE_TDM 0
#endif

#if __has_include(<hip/amd_detail/amd_gfx1250_TDM.h>)
#  define TDM_SIX_ARGS 1
#else
#  define TDM_SIX_ARGS 0
#endif

// Issue one 2-D TDM tile load: tile_d1 rows of tile_d0 fp32 elements, row
// stride strideElems, destination LDS offset ldsOff, with optional LDS row
// padding (padI/padA are the encoded D# pad_interval / pad_amount fields).
__device__ __forceinline__ void tdm_load_2d(unsigned ldsOff, const float* g,
                                            int tile_d0, int tile_d1,
                                            int strideElems, int padI, int padA) {
#if HAVE_TDM
    unsigned long long ga = (unsigned long long)(uintptr_t)g;
    u32x4 g0 = {0u, 0u, 0u, 0u};
    g0[0] = 1u;                                   // count=1, user descriptor
    g0[1] = ldsOff;                               // lds_addr (bytes)
    g0[2] = (unsigned)ga;                         // global_addr[31:0]
    g0[3] = (unsigned)((ga >> 32) & 0x01FFFFFFull) | 0x80000000u; // addr[56:32] | type=2
    unsigned u0 = 2u << 16;                       // data_size = 4 bytes
    if (padA) u0 |= (1u << 20) | ((unsigned)padI << 22) | ((unsigned)padA << 25);
    const unsigned td0 = 1u << 30, td1 = 1u << 30;  // huge tensor dims: no OOB clip
    unsigned long long st = (unsigned long long)(unsigned)strideElems;
    i32x8 g1 = {0, 0, 0, 0, 0, 0, 0, 0};
    g1[0] = (int)u0;
    g1[1] = (int)(td0 << 16);                                  // tensor_dim0[15:0]
    g1[2] = (int)((td0 >> 16) | (td1 << 16));                  // dim0 hi | dim1 lo
    g1[3] = (int)((td1 >> 16) | ((unsigned)tile_d0 << 16));    // dim1 hi | tile_dim0
    g1[4] = (int)(unsigned)tile_d1;                            // tile_dim1 (tile_dim2=0)
    g1[5] = (int)(unsigned)(st & 0xFFFFFFFFull);               // tensor_dim0_stride lo
    g1[6] = (int)(unsigned)((st >> 32) & 0xFFFFull);           // stride hi
    g1[7] = 0;
    i32x4 z4 = {0, 0, 0, 0};
#if TDM_SIX_ARGS
    i32x8 z8 = {0, 0, 0, 0, 0, 0, 0, 0};
    __builtin_amdgcn_tensor_load_to_lds(g0, g1, z4, z4, z8, 0);
#else
    __builtin_amdgcn_tensor_load_to_lds(g0, g1, z4, z4, 0);
#endif
#else
    (void)ldsOff; (void)g; (void)tile_d0; (void)tile_d1;
    (void)strideElems; (void)padI; (void)padA;
#endif
}

__device__ __forceinline__ unsigned lds_off(const void* p) {
    return (unsigned)(uintptr_t)p;   // LDS flat addr: low 32 bits = LDS byte offset
}

// ---------------------------------------------------------------------------
// Row means over T=1200 for [64*1024, 1200]; one wave32 per row.
// ---------------------------------------------------------------------------
__global__ void mean_kernel(const float* __restrict__ x, float* __restrict__ mean) {
    int wid  = blockIdx.x * 8 + (threadIdx.x >> 5);
    int lane = threadIdx.x & 31;
    const float* row = x + (size_t)wid * 1200;
    float acc = 0.f;
    for (int t = lane; t < 1200; t += 32) acc += row[t];
    #pragma unroll
    for (int off = 16; off > 0; off >>= 1) acc += __shfl_xor(acc, off, 32);
    if (lane == 0) mean[wid] = acc * (1.0f / 1200.0f);
}

// ---------------------------------------------------------------------------
// Covariance: C[b] = (xc xc^T)/1199, centering fused at fragment-read time.
// 128 threads / 4 waves; block tile 64x64, K panel 16.
// TDM double-buffered: panel i+1 DMA overlaps panel i WMMA (TENSORcnt<=2).
// ---------------------------------------------------------------------------
__global__ __launch_bounds__(128)
void cov_wmma_kernel(const float* __restrict__ x,
                     const float* __restrict__ mean,
                     float* __restrict__ C) {
    __shared__ float Xa[2][64 * 20];   // [row][k] 16+4 pad
    __shared__ float Xb[2][64 * 20];
    const int tiles = 1024 / 64;
    int b  = blockIdx.x / (tiles * tiles);
    int t  = blockIdx.x % (tiles * tiles);
    int m0 = (t / tiles) * 64;
    int n0 = (t % tiles) * 64;

    const float* xb = x + (size_t)b * 1024 * 1200;
    const float* mb = mean + (size_t)b * 1024;

    int tid  = threadIdx.x;
    int w    = tid >> 5;
    int lane = tid & 31;
    int half = lane >> 4;
    int lq   = lane & 15;
    int ml   = (w >> 1) * 32;       // wave-local 32x32 quadrant
    int nl   = (w & 1) * 32;

    float am0 = mb[m0 + ml + lq],      am1 = mb[m0 + ml + 16 + lq];
    float bm0 = mb[n0 + nl + lq],      bm1 = mb[n0 + nl + 16 + lq];

    v8f c00 = {}, c01 = {}, c10 = {}, c11 = {};

    auto compute_panel = [&](const float* Sa, const float* Sb) {
        #pragma unroll
        for (int kk = 0; kk < 16; kk += 4) {
            int ks = kk + half * 2;
            v2f a0, a1, b0, b1;
            a0[0] = Sa[(ml + lq)      * 20 + ks]     - am0;
            a0[1] = Sa[(ml + lq)      * 20 + ks + 1] - am0;
            a1[0] = Sa[(ml + 16 + lq) * 20 + ks]     - am1;
            a1[1] = Sa[(ml + 16 + lq) * 20 + ks + 1] - am1;
            b0[0] = Sb[(nl + lq)      * 20 + ks]     - bm0;
            b0[1] = Sb[(nl + lq)      * 20 + ks + 1] - bm0;
            b1[0] = Sb[(nl + 16 + lq) * 20 + ks]     - bm1;
            b1[1] = Sb[(nl + 16 + lq) * 20 + ks + 1] - bm1;
            c00 = __builtin_amdgcn_wmma_f32_16x16x4_f32(false, a0, false, b0, (short)0, c00, false, false);
            c01 = __builtin_amdgcn_wmma_f32_16x16x4_f32(false, a0, false, b1, (short)0, c01, false, false);
            c10 = __builtin_amdgcn_wmma_f32_16x16x4_f32(false, a1, false, b0, (short)0, c10, false, false);
            c11 = __builtin_amdgcn_wmma_f32_16x16x4_f32(false, a1, false, b1, (short)0, c11, false, false);
        }
    };

#if HAVE_TDM
    const int nPan = 1200 / 16;
    if (w == 0) {
        tdm_load_2d(lds_off(&Xa[0][0]), xb + (size_t)m0 * 1200, 16, 64, 1200, 3, 3);
        tdm_load_2d(lds_off(&Xb[0][0]), xb + (size_t)n0 * 1200, 16, 64, 1200, 3, 3);
    }
    for (int ip = 0; ip < nPan; ++ip) {
        if (w == 0) {
            if (ip + 1 < nPan) {
                int k1 = (ip + 1) * 16;
                tdm_load_2d(lds_off(&Xa[(ip + 1) & 1][0]), xb + (size_t)m0 * 1200 + k1, 16, 64, 1200, 3, 3);
                tdm_load_2d(lds_off(&Xb[(ip + 1) & 1][0]), xb + (size_t)n0 * 1200 + k1, 16, 64, 1200, 3, 3);
                __builtin_amdgcn_s_wait_tensorcnt(2);   // panel ip landed; ip+1 in flight
            } else {
                __builtin_amdgcn_s_wait_tensorcnt(0);
            }
        }
        __syncthreads();
        compute_panel(Xa[ip & 1], Xb[ip & 1]);
        __syncthreads();   // reads done before panel ip+2 overwrites this buffer
    }
#else
    for (int k0 = 0; k0 < 1200; k0 += 16) {
        for (int i = tid; i < 64 * 16; i += 128) {
            int r = i >> 4, k = i & 15;
            Xa[0][r * 20 + k] = xb[(size_t)(m0 + r) * 1200 + k0 + k];
            Xb[0][r * 20 + k] = xb[(size_t)(n0 + r) * 1200 + k0 + k];
        }
        __syncthreads();
        compute_panel(Xa[0], Xb[0]);
        __syncthreads();
    }
#endif

    const float alpha = 1.0f / 1199.0f;
    float* Cb = C + (size_t)b * 1024 * 1024;
    int mw = m0 + ml, nw = n0 + nl;
    #pragma unroll
    for (int v = 0; v < 8; ++v) {
        int mr = v + half * 8;
        Cb[(size_t)(mw + mr)      * 1024 + nw + lq]      = alpha * c00[v];
        Cb[(size_t)(mw + mr)      * 1024 + nw + 16 + lq] = alpha * c01[v];
        Cb[(size_t)(mw + 16 + mr) * 1024 + nw + lq]      = alpha * c10[v];
        Cb[(size_t)(mw + 16 + mr) * 1024 + nw + 16 + lq] = alpha * c11[v];
    }
}

// ---------------------------------------------------------------------------
// LDS-staged batched WMMA fp32 GEMM: C[b] = alpha * opA(A[b]) * opB(B[b]).
// TA: opA(A)[m,k] = A[k*lda+m]; TB: opB(B)[k,n] = B[n*ldb+k].
// stride{A,B} == 0 broadcasts across the batch.  Block tile 64x64, K panel 16,
// 128 threads (4 waves, each a 32x32 / 2x2-WMMA register tile).
// TDM double-buffered like the covariance kernel.
// Requires M%64==0, N%64==0, K%16==0.
// ---------------------------------------------------------------------------
template <bool TA, bool TB>
__global__ __launch_bounds__(128)
void gemm_wmma_kernel(const float* __restrict__ A,
                      const float* __restrict__ B,
                      float* __restrict__ C,
                      int M, int N, int K,
                      int lda, int ldb, int ldc,
                      int strideA, int strideB, int strideC,
                      float alpha) {
    constexpr int ASZ = TA ? 16 * 68 : 64 * 20;   // [k][m] pad4  vs  [m][k] pad4
    constexpr int BSZ = TB ? 64 * 20 : 16 * 68;   // [n][k] pad4  vs  [k][n] pad4
    __shared__ float As[2][ASZ];
    __shared__ float Bs[2][BSZ];

    int tilesN = N >> 6;
    int tilesPerBatch = (M >> 6) * tilesN;
    int b  = blockIdx.x / tilesPerBatch;
    int t  = blockIdx.x % tilesPerBatch;
    int m0 = (t / tilesN) << 6;
    int n0 = (t % tilesN) << 6;

    const float* Ab = A + (size_t)b * strideA;
    const float* Bb = B + (size_t)b * strideB;

    int tid  = threadIdx.x;
    int w    = tid >> 5;
    int lane = tid & 31;
    int half = lane >> 4;
    int lq   = lane & 15;
    int ml   = (w >> 1) * 32;
    int nl   = (w & 1) * 32;

    v8f c00 = {}, c01 = {}, c10 = {}, c11 = {};

    auto compute_panel = [&](const float* Sa, const float* Sb) {
        #pragma unroll
        for (int kk = 0; kk < 16; kk += 4) {
            int ks = kk + half * 2;
            v2f a0, a1, b0, b1;
            if (!TA) {
                a0[0] = Sa[(ml + lq)      * 20 + ks];   a0[1] = Sa[(ml + lq)      * 20 + ks + 1];
                a1[0] = Sa[(ml + 16 + lq) * 20 + ks];   a1[1] = Sa[(ml + 16 + lq) * 20 + ks + 1];
            } else {
                a0[0] = Sa[ks * 68 + ml + lq];          a0[1] = Sa[(ks + 1) * 68 + ml + lq];
                a1[0] = Sa[ks * 68 + ml + 16 + lq];     a1[1] = Sa[(ks + 1) * 68 + ml + 16 + lq];
            }
            if (!TB) {
                b0[0] = Sb[ks * 68 + nl + lq];          b0[1] = Sb[(ks + 1) * 68 + nl + lq];
                b1[0] = Sb[ks * 68 + nl + 16 + lq];     b1[1] = Sb[(ks + 1) * 68 + nl + 16 + lq];
            } else {
                b0[0] = Sb[(nl + lq)      * 20 + ks];   b0[1] = Sb[(nl + lq)      * 20 + ks + 1];
                b1[0] = Sb[(nl + 16 + lq) * 20 + ks];   b1[1] = Sb[(nl + 16 + lq) * 20 + ks + 1];
            }
            c00 = __builtin_amdgcn_wmma_f32_16x16x4_f32(false, a0, false, b0, (short)0, c00, false, false);
            c01 = __builtin_amdgcn_wmma_f32_16x16x4_f32(false, a0, false, b1, (short)0, c01, false, false);
            c10 = __builtin_amdgcn_wmma_f32_16x16x4_f32(false, a1, false, b0, (short)0, c10, false, false);
            c11 = __builtin_amdgcn_wmma_f32_16x16x4_f32(false, a1, false, b1, (short)0, c11, false, false);
        }
    };

#if HAVE_TDM
    auto issue_panel = [&](int k0, int bi) {
        if (!TA) tdm_load_2d(lds_off(&As[bi][0]), Ab + (size_t)m0 * lda + k0, 16, 64, lda, 3, 3);
        else     tdm_load_2d(lds_off(&As[bi][0]), Ab + (size_t)k0 * lda + m0, 64, 16, lda, 5, 3);
        if (!TB) tdm_load_2d(lds_off(&Bs[bi][0]), Bb + (size_t)k0 * ldb + n0, 64, 16, ldb, 5, 3);
        else     tdm_load_2d(lds_off(&Bs[bi][0]), Bb + (size_t)n0 * ldb + k0, 16, 64, ldb, 3, 3);
    };
    const int nPan = K >> 4;
    if (w == 0) issue_panel(0, 0);
    for (int ip = 0; ip < nPan; ++ip) {
        if (w == 0) {
            if (ip + 1 < nPan) {
                issue_panel((ip + 1) << 4, (ip + 1) & 1);
                __builtin_amdgcn_s_wait_tensorcnt(2);
            } else {
                __builtin_amdgcn_s_wait_tensorcnt(0);
            }
        }
        __syncthreads();
        compute_panel(As[ip & 1], Bs[ip & 1]);
        __syncthreads();
    }
#else
    for (int k0 = 0; k0 < K; k0 += 16) {
        if (!TA) {
            for (int i = tid; i < 64 * 16; i += 128) {
                int r = i >> 4, k = i & 15;
                As[0][r * 20 + k] = Ab[(size_t)(m0 + r) * lda + k0 + k];
            }
        } else {
            for (int i = tid; i < 16 * 64; i += 128) {
                int r = i >> 6, m = i & 63;
                As[0][r * 68 + m] = Ab[(size_t)(k0 + r) * lda + m0 + m];
            }
        }
        if (!TB) {
            for (int i = tid; i < 16 * 64; i += 128) {
                int r = i >> 6, n = i & 63;
                Bs[0][r * 68 + n] = Bb[(size_t)(k0 + r) * ldb + n0 + n];
            }
        } else {
            for (int i = tid; i < 64 * 16; i += 128) {
                int r = i >> 4, k = i & 15;
                Bs[0][r * 20 + k] = Bb[(size_t)(n0 + r) * ldb + k0 + k];
            }
        }
        __syncthreads();
        compute_panel(As[0], Bs[0]);
        __syncthreads();
    }
#endif

    float* Cb = C + (size_t)b * strideC;
    int mw = m0 + ml, nw = n0 + nl;
    #pragma unroll
    for (int v = 0; v < 8; ++v) {
        int mr = v + half * 8;
        Cb[(size_t)(mw + mr)      * ldc + nw + lq]      = alpha * c00[v];
        Cb[(size_t)(mw + mr)      * ldc + nw + 16 + lq] = alpha * c01[v];
        Cb[(size_t)(mw + 16 + mr) * ldc + nw + lq]      = alpha * c10[v];
        Cb[(size_t)(mw + 16 + mr) * ldc + nw + 16 + lq] = alpha * c11[v];
    }
}

// ---------------------------------------------------------------------------
// Batched parallel cyclic Jacobi eigensolver, one workgroup per matrix.
// ---------------------------------------------------------------------------
template <int N>
__global__ void jacobi_kernel(float* __restrict__ Abuf, float* __restrict__ Vbuf) {
    __shared__ int   ord[N];
    __shared__ float cs[N / 2];
    __shared__ float sn[N / 2];
    float* A = Abuf + (size_t)blockIdx.x * N * N;
    float* V = Vbuf + (size_t)blockIdx.x * N * N;
    const int tid = threadIdx.x;
    const int nt  = blockDim.x;

    for (int idx = tid; idx < N * N; idx += nt) {
        int i = idx / N, j = idx - i * N;
        V[idx] = (i == j) ? 1.0f : 0.0f;
    }
    for (int i = tid; i < N; i += nt) ord[i] = i;
    __syncthreads();

    for (int sweep = 0; sweep < NSWEEP; ++sweep) {
        for (int step = 0; step < N - 1; ++step) {
            for (int k = tid; k < N / 2; k += nt) {
                int p = ord[k], q = ord[N - 1 - k];
                if (p > q) { int tt = p; p = q; q = tt; }
                float app = A[(size_t)p * N + p];
                float aqq = A[(size_t)q * N + q];
                float apq = A[(size_t)p * N + q];
                float c = 1.0f, s = 0.0f;
                if (fabsf(apq) > 1e-20f) {
                    float tau = (aqq - app) / (2.0f * apq);
                    float tt  = ((tau >= 0.0f) ? 1.0f : -1.0f) /
                                (fabsf(tau) + sqrtf(1.0f + tau * tau));
                    c = rsqrtf(1.0f + tt * tt);
                    s = tt * c;
                }
                cs[k] = c; sn[k] = s;
            }
            __syncthreads();
            for (int idx = tid; idx < (N / 2) * N; idx += nt) {   // rows: A <- J^T A
                int k = idx / N, j = idx - k * N;
                int p = ord[k], q = ord[N - 1 - k];
                if (p > q) { int tt = p; p = q; q = tt; }
                float c = cs[k], s = sn[k];
                float ap = A[(size_t)p * N + j], aq = A[(size_t)q * N + j];
                A[(size_t)p * N + j] = c * ap - s * aq;
                A[(size_t)q * N + j] = s * ap + c * aq;
            }
            __syncthreads();
            for (int idx = tid; idx < (N / 2) * N; idx += nt) {   // cols: A <- A J; V <- V J
                int k = idx / N, i = idx - k * N;
                int p = ord[k], q = ord[N - 1 - k];
                if (p > q) { int tt = p; p = q; q = tt; }
                float c = cs[k], s = sn[k];
                float ap = A[(size_t)i * N + p], aq = A[(size_t)i * N + q];
                A[(size_t)i * N + p] = c * ap - s * aq;
                A[(size_t)i * N + q] = s * ap + c * aq;
                float vp = V[(size_t)i * N + p], vq = V[(size_t)i * N + q];
                V[(size_t)i * N + p] = c * vp - s * vq;
                V[(size_t)i * N + q] = s * vp + c * vq;
            }
            __syncthreads();
            if (tid == 0) {
                int last = ord[N - 1];
                for (int i = N - 1; i > 1; --i) ord[i] = ord[i - 1];
                ord[1] = last;
            }
            __syncthreads();
        }
    }
}

// ---------------------------------------------------------------------------
// B[b][i][j] = V[b][i][j] * f(diag(A[b])[j]) ; f = max(.,eps) or log.
// ---------------------------------------------------------------------------
__global__ void scale_cols_kernel(const float* __restrict__ A,
                                  const float* __restrict__ V,
                                  float* __restrict__ B,
                                  int n, int useLog, long long total) {
    long long stride = (long long)gridDim.x * blockDim.x;
    int nn = n * n;
    for (long long idx = (long long)blockIdx.x * blockDim.x + threadIdx.x;
         idx < total; idx += stride) {
        int b = (int)(idx / nn);
        int r = (int)(idx - (long long)b * nn);
        int j = r % n;
        float d = A[(size_t)b * nn + (size_t)j * n + j];
        float f = useLog ? logf(fmaxf(d, 1e-30f)) : fmaxf(d, REEIG_EPS);
        B[idx] = V[idx] * f;
    }
}

// ---------------------------------------------------------------------------
// out[b,c] = dot(X[b,:4096], Wl[c,:4096]) + bias[c]; one wave per output.
// ---------------------------------------------------------------------------
__global__ void linear_kernel(const float* __restrict__ X,
                              const float* __restrict__ Wl,
                              const float* __restrict__ bias,
                              float* __restrict__ out) {
    int wid = blockIdx.x * 8 + (threadIdx.x >> 5);
    if (wid >= 64 * 11) return;
    int lane = threadIdx.x & 31;
    int b = wid / 11, c = wid % 11;
    const float* xv = X  + (size_t)b * 4096;
    const float* wv = Wl + (size_t)c * 4096;
    float acc = 0.f;
    for (int i = lane; i < 4096; i += 32) acc += xv[i] * wv[i];
    #pragma unroll
    for (int off = 16; off > 0; off >>= 1) acc += __shfl_xor(acc, off, 32);
    if (lane == 0) out[(size_t)b * 11 + c] = acc + bias[c];
}

// ---------------------------------------------------------------------------
// Host orchestration
// ---------------------------------------------------------------------------
extern "C" void kernel_launch(void* const* d_in, const int* in_sizes, int n_in,
                              void* d_out, int out_size, void* d_ws, size_t ws_size,
                              hipStream_t stream) {
    const float* x     = (const float*)d_in[0];   // [64,1024,1200]
    const float* Ws[4] = {(const float*)d_in[1], (const float*)d_in[2],
                          (const float*)d_in[3], (const float*)d_in[4]};
    const float* lin_w = (const float*)d_in[5];   // [11,4096]
    const float* lin_b = (const float*)d_in[6];   // [11]
    float* out = (float*)d_out;                   // [64,11]

    char* ws = (char*)d_ws;
    const size_t offX = 1ull << 20;
    const size_t offT = offX + 64ull * 1024 * 1024 * 4;   // X: 268.4 MB
    const size_t offY = offT + 64ull * 1024 * 512 * 4;    // T: 134.2 MB
    float* mean = (float*)ws;
    float* bufX = (float*)(ws + offX);      // current SPD matrices
    float* bufT = (float*)(ws + offT);      // bimap temp; aliased by V later
    float* bufY = (float*)(ws + offY);      // bimap output / jacobi in-place
    float* bufV = bufT;                     // eigenvectors (T dead by then)
    float* bufB = bufT + 64ull * 512 * 512; // scaled eigenvectors

    // 1) row means
    mean_kernel<<<(64 * 1024) / 8, 256, 0, stream>>>(x, mean);

    // 2) covariance pooling (fused centering): X = cov  [64,1024,1024]
    cov_wmma_kernel<<<64 * 16 * 16, 128, 0, stream>>>(x, mean, bufX);

    // 3) four stages: BiMap -> Jacobi eigh -> ReEig/LogEig reconstruct
    const int dims[5] = {1024, 512, 256, 128, 64};
    for (int s = 0; s < 4; ++s) {
        int ni = dims[s], no = dims[s + 1];

        // T[b] (ni x no) = X[b] * W
        gemm_wmma_kernel<false, false><<<64 * (ni / 64) * (no / 64), 128, 0, stream>>>(
            bufX, Ws[s], bufT, ni, no, ni, ni, no, no,
            ni * ni, 0, ni * no, 1.0f);

        // Y[b] (no x no) = W^T * T[b]
        gemm_wmma_kernel<true, false><<<64 * (no / 64) * (no / 64), 128, 0, stream>>>(
            Ws[s], bufT, bufY, no, no, ni, no, no, no,
            0, ni * no, no * no, 1.0f);

        // batched Jacobi eigendecomposition: Y -> diag(evals), V = evecs
        switch (no) {
            case 512: jacobi_kernel<512><<<64, 256, 0, stream>>>(bufY, bufV); break;
            case 256: jacobi_kernel<256><<<64, 256, 0, stream>>>(bufY, bufV); break;
            case 128: jacobi_kernel<128><<<64, 256, 0, stream>>>(bufY, bufV); break;
            default:  jacobi_kernel< 64><<<64, 256, 0, stream>>>(bufY, bufV); break;
        }

        // B = V * diag(f(evals))
        long long total = 64ll * no * no;
        scale_cols_kernel<<<2048, 256, 0, stream>>>(bufY, bufV, bufB, no,
                                                    (s == 3) ? 1 : 0, total);

        // X_next[b] = B[b] * V[b]^T
        gemm_wmma_kernel<false, true><<<64 * (no / 64) * (no / 64), 128, 0, stream>>>(
            bufB, bufV, bufX, no, no, no, no, no, no,
            no * no, no * no, no * no, 1.0f);
    }

    // 4) final linear head
    linear_kernel<<<(64 * 11 + 7) / 8, 256, 0, stream>>>(bufX, lin_w, lin_b, out);
}